// CompoundLoss_20358144983671
// MI455X (gfx1250) — compile-verified
//
#include <hip/hip_runtime.h>
#include <cstdint>
#include <cstddef>

// Problem constants (from reference)
#define NB 2
#define ND 128
#define NH 256
#define NW 256
#define NWW 8                        // words along W (256/32)
#define NTOT (NB*ND*NH*NW)           // 16,777,216 voxels
#define NPACK (NTOT/32)              // 524,288 packed words (2MB)

// Main-kernel tiling: each block handles ZT z-slices x YT rows x full W
#define ZT 4
#define YT 32
#define ZL (ZT+6)                    // 10 staged z-levels (halo 3)
#define YL (YT+6)                    // 38 staged y-rows  (halo 3)
#define TILE_WORDS (ZL*YL*NWW)       // 3040 words = 12160 B LDS

typedef float v2f __attribute__((ext_vector_type(2)));
typedef float v8f __attribute__((ext_vector_type(8)));

// ---------------------------------------------------------------------------
// Pack targets (f32 0/1) into bits, 32 voxels per uint32 along W.
// ---------------------------------------------------------------------------
__global__ __launch_bounds__(256) void pack_kernel(const float* __restrict__ targets,
                                                   uint32_t* __restrict__ pack) {
  uint32_t g = blockIdx.x * 256u + threadIdx.x;
  bool fg = targets[g] > 0.5f;
#if __has_builtin(__builtin_amdgcn_ballot_w32)
  uint32_t m = __builtin_amdgcn_ballot_w32(fg);
#else
  uint32_t m = (uint32_t)__ballot(fg);
#endif
  if ((threadIdx.x & 31u) == 0u) pack[g >> 5] = m;
}

// ---------------------------------------------------------------------------
// Exact dual 32-lane sum with ONE V_WMMA_F32_16X16X4_F32, then a single
// ds_swizzle (SWAPX16) cross-half add, then predicated f64 global atomics.
//   A: vgpr0 = x -> A[m, K even], vgpr1 = y -> A[m, K odd].
//   B[k,n] = 1 iff (k even) == (n<8)  => lanes (<8 | >=24) hold 1.0.
//   D[m, n<8] = x_m + x_{m+16}; D[m, n>=8] = y_m + y_{m+16}.
//   After per-lane 8-add + swapx16-add: lane 0 = total(x), lane 8 = total(y).
// ---------------------------------------------------------------------------
__device__ __forceinline__ void wave_sum_pair_atomic(float x, float y,
                                                     double* __restrict__ acc, int j) {
  int lane = threadIdx.x & 31;
  float bsel = ((lane < 8) || (lane >= 24)) ? 1.0f : 0.0f;
  v2f a; a[0] = x;    a[1] = y;
  v2f b; b[0] = bsel; b[1] = bsel;
  v8f c = {};
  v8f d = __builtin_amdgcn_wmma_f32_16x16x4_f32(false, a, false, b,
                                                (short)0, c, false, false);
  float h = ((d[0]+d[1])+(d[2]+d[3])) + ((d[4]+d[5])+(d[6]+d[7]));
  // SWAPX16: group-of-32 swizzle, xor_mask=0x10, and_mask=0x1f
  h += __int_as_float(__builtin_amdgcn_ds_swizzle(__float_as_int(h), 0x401F));
  if ((lane & 0x17) == 0)                     // lanes 0 and 8
    atomicAdd(&acc[j + (lane >> 3)], (double)h);
}

// ---------------------------------------------------------------------------
// Fused main kernel: async-stage packed tile to LDS, bitwise Chebyshev
// distance (radii 1..3, clamp 4), pointwise BCE/sigmoid terms, 6 sums.
// acc layout: [0]=sum(v) [1]=sum(bce*v) [2]=sum(p*t*v) [3]=sum(p*v)
//             [4]=sum(t*v) [5]=sum(p*signed_dist*v)
// ---------------------------------------------------------------------------
__global__ __launch_bounds__(256)
__attribute__((amdgpu_waves_per_eu(4)))
void main_kernel(const float* __restrict__ logits,
                 const float* __restrict__ valid,
                 const uint32_t* __restrict__ pack,
                 double* __restrict__ acc) {
  __shared__ uint32_t tile[ZL][YL][NWW];

  const int tid = threadIdx.x;
  const int bid = blockIdx.x;
  const int b   = bid / ((ND/ZT)*(NH/YT));
  const int r   = bid % ((ND/ZT)*(NH/YT));
  const int z0  = (r / (NH/YT)) * ZT;
  const int y0  = (r % (NH/YT)) * YT;

  // ---- Stage packed tile into LDS with CDNA5 async loads ----
  const uint32_t* packB = pack + (size_t)b * (ND*NH*NWW);
  for (int idx = tid; idx < TILE_WORDS; idx += 256) {
    int li  = idx / (YL*NWW);
    int rr  = idx - li*(YL*NWW);
    int yi  = rr / NWW;
    int wwi = rr - yi*NWW;
    int gz = z0 - 3 + li;
    int gy = y0 - 3 + yi;
    if (gz >= 0 && gz < ND && gy >= 0 && gy < NH) {
      const uint32_t* src = packB + ((size_t)gz*NH + gy)*NWW + wwi;
      uint32_t lds_off = (uint32_t)(uintptr_t)&tile[li][yi][wwi];
      uint64_t ga = (uint64_t)(uintptr_t)src;
      asm volatile("global_load_async_to_lds_b32 %0, %1, off"
                   :: "v"(lds_off), "v"(ga) : "memory");
    }
  }
  asm volatile("s_wait_asynccnt 0" ::: "memory");
  __syncthreads();

  const int yl = tid >> 3;   // 0..31
  const int ww = tid & 7;    // 0..7
  const int y  = y0 + yl;

  // Window OR (max) / AND (min) words per output z, radii 1..3.
  uint32_t M1[ZT], M2[ZT], M3[ZT], N1[ZT], N2[ZT], N3[ZT];

  {
    // Per-z-level, y-combined words (live only inside this scope).
    uint32_t YO1[ZL], YO2[ZL], YO3[ZL], YA1[ZL], YA2[ZL], YA3[ZL];

#pragma unroll
    for (int li = 0; li < ZL; ++li) {
      int gz = z0 - 3 + li;
      uint32_t o1 = 0, o2 = 0, o3 = 0;
      uint32_t a1 = 0xFFFFFFFFu, a2 = 0xFFFFFFFFu, a3 = 0xFFFFFFFFu;
      if (gz >= 0 && gz < ND) {
#pragma unroll
        for (int dy = -3; dy <= 3; ++dy) {
          int gy = y + dy;
          if (gy < 0 || gy >= NH) continue;   // SAME padding: identity
          int yi = yl + 3 + dy;
          uint32_t cur   = tile[li][yi][ww];
          uint32_t prevw = (ww > 0) ? tile[li][yi][ww-1] : 0u;
          uint32_t nextw = (ww < 7) ? tile[li][yi][ww+1] : 0u;
          // 38-bit row window: bit (i+3) = voxel i, i in [-3,34]
          uint64_t Ro = ((uint64_t)prevw >> 29) | ((uint64_t)cur << 3) |
                        ((uint64_t)(nextw & 7u) << 35);
          uint64_t Ra = Ro;
          if (ww == 0) Ra |= 7ull;            // W-edge: AND identity = 1
          if (ww == 7) Ra |= (7ull << 35);
          // span OR: S_n[b] = OR R[b..b+n-1]
          uint64_t t2 = Ro | (Ro >> 1);
          uint64_t s3 = t2 | (Ro >> 2);
          uint64_t t4 = t2 | (t2 >> 2);
          uint64_t s5 = t4 | (Ro >> 4);
          uint64_t s7 = t4 | (t4 >> 3);
          // span AND
          uint64_t u2 = Ra & (Ra >> 1);
          uint64_t q3 = u2 & (Ra >> 2);
          uint64_t u4 = u2 & (u2 >> 2);
          uint64_t q5 = u4 & (Ra >> 4);
          uint64_t q7 = u4 & (u4 >> 3);
          uint32_t wo1 = (uint32_t)(s3 >> 2), wo2 = (uint32_t)(s5 >> 1), wo3 = (uint32_t)s7;
          uint32_t wa1 = (uint32_t)(q3 >> 2), wa2 = (uint32_t)(q5 >> 1), wa3 = (uint32_t)q7;
          int ad = dy < 0 ? -dy : dy;
          if (ad <= 1) { o1 |= wo1; a1 &= wa1; }
          if (ad <= 2) { o2 |= wo2; a2 &= wa2; }
          o3 |= wo3; a3 &= wa3;
        }
      }
      YO1[li] = o1; YO2[li] = o2; YO3[li] = o3;
      YA1[li] = a1; YA2[li] = a2; YA3[li] = a3;
    }

    // Combine along z now so the big arrays die before the pointwise loop.
#pragma unroll
    for (int zo = 0; zo < ZT; ++zo) {
      M1[zo] = YO1[zo+2] | YO1[zo+3] | YO1[zo+4];
      M2[zo] = YO2[zo+1] | YO2[zo+2] | YO2[zo+3] | YO2[zo+4] | YO2[zo+5];
      M3[zo] = YO3[zo]   | YO3[zo+1] | YO3[zo+2] | YO3[zo+3] | YO3[zo+4] | YO3[zo+5] | YO3[zo+6];
      N1[zo] = YA1[zo+2] & YA1[zo+3] & YA1[zo+4];
      N2[zo] = YA2[zo+1] & YA2[zo+2] & YA2[zo+3] & YA2[zo+4] & YA2[zo+5];
      N3[zo] = YA3[zo]   & YA3[zo+1] & YA3[zo+2] & YA3[zo+3] & YA3[zo+4] & YA3[zo+5] & YA3[zo+6];
    }
  }

  float s_v = 0.f, s_bce = 0.f, s_i = 0.f, s_sp = 0.f, s_st = 0.f, s_bn = 0.f;

#pragma unroll
  for (int zo = 0; zo < ZT; ++zo) {
    uint32_t m1 = M1[zo], m2 = M2[zo], m3 = M3[zo];
    uint32_t n1 = N1[zo], n2 = N2[zo], n3 = N3[zo];
    uint32_t tw = tile[zo+3][yl+3][ww];

    size_t base = ((((size_t)b*ND + (z0+zo))*NH + y)*NW) + (size_t)ww*32;
    const float4* lp = (const float4*)(logits + base);
    const float4* vp = (const float4*)(valid  + base);
#pragma unroll 2
    for (int q = 0; q < 8; ++q) {
      float4 L = lp[q];
      float4 V = vp[q];
      float xs[4] = {L.x, L.y, L.z, L.w};
      float vs[4] = {V.x, V.y, V.z, V.w};
#pragma unroll
      for (int j = 0; j < 4; ++j) {
        int i = (q & 1)*4 + j;      // unused; bit position within tw is sh
        uint32_t sh = (uint32_t)(q*4 + j);
        float x  = xs[j];
        float vm = vs[j];
        float tb = (float)((tw >> sh) & 1u);
        float e  = __expf(-fabsf(x));
        float lg = __logf(1.0f + e);                 // log1p(exp(-|x|))
        float bce = fmaxf(x, 0.0f) - x*tb + lg;
        float rc = __builtin_amdgcn_rcpf(1.0f + e);
        float p  = (x >= 0.0f) ? rc : e * rc;        // sigmoid(x)
        float dn = 4.0f - (float)((m1>>sh)&1u) - (float)((m2>>sh)&1u) - (float)((m3>>sh)&1u);
        float dp = 1.0f + (float)((n1>>sh)&1u) + (float)((n2>>sh)&1u) + (float)((n3>>sh)&1u);
        float sg = (tb > 0.0f) ? -dp : dn;           // dn_b - dp_b (signed)
        float pv = p * vm;
        s_v   += vm;
        s_bce += bce * vm;
        s_sp  += pv;
        s_i   += pv * tb;
        s_st  += vm * tb;
        s_bn  += pv * sg;
        (void)i;
      }
    }
  }

  // ---- reduction: 3 WMMAs per wave (2 sums each) -> f64 global atomics ----
  wave_sum_pair_atomic(s_v,  s_bce, acc, 0);
  wave_sum_pair_atomic(s_i,  s_sp,  acc, 2);
  wave_sum_pair_atomic(s_st, s_bn,  acc, 4);
}

// ---------------------------------------------------------------------------
// Finalize: combine 6 sums into (total, bce, surface_dice, boundary).
// ---------------------------------------------------------------------------
__global__ void fin_kernel(const double* __restrict__ acc, float* __restrict__ out) {
  double vs  = acc[0] > 1.0 ? acc[0] : 1.0;
  double bce = acc[1] / vs;
  double I   = acc[2];
  double U   = acc[3] + acc[4];
  double dice = 1.0 - (2.0*I + 1e-5) / (U + 1e-5);
  double boundary = acc[5] / vs;
  double total = bce + dice + 0.5*boundary;
  out[0] = (float)total;
  out[1] = (float)bce;
  out[2] = (float)dice;
  out[3] = (float)boundary;
}

extern "C" void kernel_launch(void* const* d_in, const int* in_sizes, int n_in,
                              void* d_out, int out_size, void* d_ws, size_t ws_size,
                              hipStream_t stream) {
  (void)in_sizes; (void)n_in; (void)out_size; (void)ws_size;
  const float* logits  = (const float*)d_in[0];
  const float* targets = (const float*)d_in[1];
  const float* valid   = (const float*)d_in[2];
  double*   acc  = (double*)d_ws;
  uint32_t* pack = (uint32_t*)((char*)d_ws + 256);

  hipMemsetAsync(acc, 0, 6*sizeof(double), stream);
  pack_kernel<<<NTOT/256, 256, 0, stream>>>(targets, pack);
  main_kernel<<<NB*(ND/ZT)*(NH/YT), 256, 0, stream>>>(logits, valid, pack, acc);
  fin_kernel<<<1, 1, 0, stream>>>(acc, (float*)d_out);
}